// STTransformerBlock_12687333392758
// MI455X (gfx1250) — compile-verified
//
// STTransformerBlock for MI455X (gfx1250), fp32 WMMA path.
//
// Roofline: ~13 GFLOP fp32; working set ~70 MB << 192 MB L2; HBM (23.3 TB/s)
// irrelevant after first touch -> compute bound -> all GEMMs via
// V_WMMA_F32_16X16X4_F32. Wide GEMMs use 16x64 register blocking per wave
// (4 WMMAs per A-fragment load -> 2.5 dwords of cache traffic per WMMA
// instead of 4). Spatial attention is flash-style (no [N,N,T,H] score
// tensor). Temporal attention (T=12, ~1% of FLOPs) stays VALU.

#include <hip/hip_runtime.h>

typedef float v2f __attribute__((ext_vector_type(2)));
typedef float v8f __attribute__((ext_vector_type(8)));

#define N_NODES   1200
#define T_STEPS   12
#define C_DIM     64
#define H_HEADS   4
#define HD        16
#define FC_DIM    256
#define C2_DIM    128
#define TIME_NUM_ 288
#define NT        (N_NODES * T_STEPS)        // 14400
#define NTC       (NT * C_DIM)               // 921600
#define ROWSTRIDE (T_STEPS * C_DIM)          // 768 floats per node row

// ---------------- reductions (wave32) ----------------
__device__ __forceinline__ float red_sum32(float v) {
  #pragma unroll
  for (int m = 16; m > 0; m >>= 1) v += __shfl_xor(v, m, 32);
  return v;
}
__device__ __forceinline__ float red_max16(float v) {
  #pragma unroll
  for (int m = 8; m > 0; m >>= 1) v = fmaxf(v, __shfl_xor(v, m, 16));
  return v;
}
__device__ __forceinline__ float red_sum16(float v) {
  #pragma unroll
  for (int m = 8; m > 0; m >>= 1) v += __shfl_xor(v, m, 16);
  return v;
}

// ---------------- wide fp32 WMMA GEMM: 16x64 strip per wave ----------------
// D[M,Nc] = A[M,K] @ B[K,Nc] (+ bias[col % bias_mod]) (optional ReLU).
// Requires M%16==0, Nc%64==0, K%4==0. One A fragment feeds 4 WMMAs.
__global__ void gemm64_wmma(const float* __restrict__ A,
                            const float* __restrict__ B,
                            const float* __restrict__ bias,
                            float* __restrict__ D,
                            int M, int K, int Nc, int bias_mod, int relu) {
  int wave  = blockIdx.x * (blockDim.x >> 5) + (threadIdx.x >> 5);
  int tn    = Nc >> 6;                 // 64-wide column strips
  int tiles = (M >> 4) * tn;
  if (wave >= tiles) return;
  int m0   = (wave / tn) << 4;
  int n0   = (wave % tn) << 6;
  int lane = threadIdx.x & 31;
  int sub  = lane >> 4;                // lane half (0/1)
  int lcol = lane & 15;

  const float* arow = A + (size_t)(m0 + lcol) * K;
  v8f acc[4];
  #pragma unroll
  for (int c = 0; c < 4; ++c) acc[c] = (v8f){0.f,0.f,0.f,0.f,0.f,0.f,0.f,0.f};

  for (int k0 = 0; k0 < K; k0 += 4) {
    // A 16x4 layout: lanes 0-15/16-31 -> rows 0-15; VGPR0=K{0,2}, VGPR1=K{1,3}
    v2f a;
    a.x = arow[k0 + 2 * sub];
    a.y = arow[k0 + 2 * sub + 1];
    __builtin_prefetch(arow + k0 + 64, 0, 0);                // global_prefetch_b8
    const float* brow = B + (size_t)(k0 + 2 * sub) * Nc + n0 + lcol;
    __builtin_prefetch(brow + (size_t)16 * Nc, 0, 0);
    #pragma unroll
    for (int c = 0; c < 4; ++c) {
      // B 4x16 layout: VGPR0 rows {0,2}, VGPR1 rows {1,3}; cols striped on lanes
      v2f b;
      b.x = brow[16 * c];
      b.y = brow[16 * c + Nc];
      acc[c] = __builtin_amdgcn_wmma_f32_16x16x4_f32(false, a, false, b,
                                                     (short)0, acc[c], false, false);
    }
  }

  // C/D layout: VGPR j -> row j (lanes 0-15) / row j+8 (lanes 16-31)
  #pragma unroll
  for (int c = 0; c < 4; ++c) {
    int col = n0 + 16 * c + lcol;
    float bv = bias ? bias[col % bias_mod] : 0.f;
    #pragma unroll
    for (int j = 0; j < 8; ++j) {
      int row = m0 + j + 8 * sub;
      float v = acc[c][j] + bv;
      if (relu) v = fmaxf(v, 0.f);
      D[(size_t)row * Nc + col] = v;
    }
  }
}

// ---------------- narrow fp32 WMMA GEMM: one 16x16 tile per wave ----------
// Used for the per-head Hd x Hd projections (Nc == 16).
__global__ void gemm16_wmma(const float* __restrict__ A,
                            const float* __restrict__ B,
                            const float* __restrict__ bias,
                            float* __restrict__ D,
                            int M, int K, int Nc, int bias_mod, int relu) {
  int wave  = blockIdx.x * (blockDim.x >> 5) + (threadIdx.x >> 5);
  int tn    = Nc >> 4;
  int tiles = (M >> 4) * tn;
  if (wave >= tiles) return;
  int m0   = (wave / tn) << 4;
  int n0   = (wave % tn) << 4;
  int lane = threadIdx.x & 31;
  int sub  = lane >> 4;
  int lcol = lane & 15;

  const float* arow = A + (size_t)(m0 + lcol) * K;
  v8f acc = {0.f, 0.f, 0.f, 0.f, 0.f, 0.f, 0.f, 0.f};

  for (int k0 = 0; k0 < K; k0 += 4) {
    v2f a, b;
    a.x = arow[k0 + 2 * sub];
    a.y = arow[k0 + 2 * sub + 1];
    const float* bp = B + (size_t)(k0 + 2 * sub) * Nc + n0 + lcol;
    b.x = bp[0];
    b.y = bp[Nc];
    acc = __builtin_amdgcn_wmma_f32_16x16x4_f32(false, a, false, b,
                                                (short)0, acc, false, false);
  }

  #pragma unroll
  for (int j = 0; j < 8; ++j) {
    int row = m0 + j + 8 * sub;
    int col = n0 + lcol;
    float v = acc[j];
    if (bias) v += bias[col % bias_mod];
    if (relu) v = fmaxf(v, 0.f);
    D[(size_t)row * Nc + col] = v;
  }
}

// ---------------- spatial flash attention ----------------
// Q,K,V,O: [N, T, H, Hd] fp32. One wave per (t, h, 16-query tile); streams
// 75 key tiles of 16. Scores via WMMA, streaming softmax over k (axis=1 in
// the reference), PV via WMMA. 1/sqrt(C)=1/8 folded into Q.
__global__ void spatial_flash_attn(const float* __restrict__ Qm,
                                   const float* __restrict__ Km,
                                   const float* __restrict__ Vm,
                                   float* __restrict__ Om) {
  __shared__ float P[16][16];   // one wave per block
  const int qtiles = N_NODES / 16;   // 75
  int bid = blockIdx.x;
  int qt  = bid % qtiles;
  int th  = bid / qtiles;            // 0..47
  int t   = th >> 2;
  int h   = th & 3;
  int off = t * C_DIM + h * HD;      // offset inside a node row
  int lane = threadIdx.x & 31;
  int sub  = lane >> 4;
  int lcol = lane & 15;
  int q0   = qt * 16;
  const float inv_scale = 0.125f;    // 1/sqrt(64)

  // Q tile (16 queries x 16 d) in A layout, pre-scaled; 4 K-steps of 4
  v2f qa[4];
  #pragma unroll
  for (int s = 0; s < 4; ++s) {
    const float* qp = Qm + (size_t)(q0 + lcol) * ROWSTRIDE + off + 4 * s + 2 * sub;
    qa[s].x = qp[0] * inv_scale;
    qa[s].y = qp[1] * inv_scale;
  }

  float mrow[8], lrow[8];
  #pragma unroll
  for (int j = 0; j < 8; ++j) { mrow[j] = -1e30f; lrow[j] = 0.f; }
  v8f acc = {0.f, 0.f, 0.f, 0.f, 0.f, 0.f, 0.f, 0.f};

  for (int kt = 0; kt < qtiles; ++kt) {
    int k0 = kt * 16;
    // scores tile S[16q x 16k] = (Q/8) @ K^T
    v8f sc = {0.f, 0.f, 0.f, 0.f, 0.f, 0.f, 0.f, 0.f};
    #pragma unroll
    for (int s = 0; s < 4; ++s) {
      v2f b;  // B[d, key] = K[key, d]
      const float* kp = Km + (size_t)(k0 + lcol) * ROWSTRIDE + off + 4 * s + 2 * sub;
      b.x = kp[0];
      b.y = kp[1];
      sc = __builtin_amdgcn_wmma_f32_16x16x4_f32(false, qa[s], false, b,
                                                 (short)0, sc, false, false);
    }
    // streaming softmax; row j+8*sub lives across 16 lanes of element j
    #pragma unroll
    for (int j = 0; j < 8; ++j) {
      float rmax  = red_max16(sc[j]);
      float mnew  = fmaxf(mrow[j], rmax);
      float alpha = __expf(mrow[j] - mnew);
      float p     = __expf(sc[j] - mnew);
      float rsum  = red_sum16(p);
      lrow[j] = lrow[j] * alpha + rsum;
      mrow[j] = mnew;
      acc[j] *= alpha;
      P[j + 8 * sub][lcol] = p;      // C-layout -> LDS (row-major tile)
    }
    // PV: A = P (rows=queries, K=key-sub), B = V (rows=key-sub, cols=d)
    #pragma unroll
    for (int s = 0; s < 4; ++s) {
      v2f a, b;
      a.x = P[lcol][4 * s + 2 * sub];
      a.y = P[lcol][4 * s + 2 * sub + 1];
      const float* vp = Vm + (size_t)(k0 + 4 * s + 2 * sub) * ROWSTRIDE + off + lcol;
      b.x = vp[0];
      b.y = vp[ROWSTRIDE];
      acc = __builtin_amdgcn_wmma_f32_16x16x4_f32(false, a, false, b,
                                                  (short)0, acc, false, false);
    }
  }

  #pragma unroll
  for (int j = 0; j < 8; ++j)
    Om[(size_t)(q0 + j + 8 * sub) * ROWSTRIDE + off + lcol] = acc[j] / lrow[j];
}

// ---------------- temporal attention (T=12, VALU) ----------------
__global__ void temporal_attn(const float* __restrict__ q2,
                              const float* __restrict__ k2,
                              const float* __restrict__ v2,
                              float* __restrict__ o) {
  int idx = blockIdx.x * blockDim.x + threadIdx.x;
  if (idx >= N_NODES * H_HEADS * T_STEPS) return;
  int qi = idx % T_STEPS;
  int nh = idx / T_STEPS;
  int h  = nh % H_HEADS;
  int n  = nh / H_HEADS;
  const float* qp = q2 + (size_t)n * ROWSTRIDE + qi * C_DIM + h * HD;
  float sc[T_STEPS];
  float mmax = -1e30f;
  #pragma unroll
  for (int k = 0; k < T_STEPS; ++k) {
    const float* kp = k2 + (size_t)n * ROWSTRIDE + k * C_DIM + h * HD;
    float d = 0.f;
    #pragma unroll
    for (int c = 0; c < HD; ++c) d += qp[c] * kp[c];
    d *= 0.125f;
    sc[k] = d;
    mmax = fmaxf(mmax, d);
  }
  float ssum = 0.f;
  #pragma unroll
  for (int k = 0; k < T_STEPS; ++k) { sc[k] = __expf(sc[k] - mmax); ssum += sc[k]; }
  float inv = 1.f / ssum;
  float out[HD];
  #pragma unroll
  for (int c = 0; c < HD; ++c) out[c] = 0.f;
  #pragma unroll
  for (int k = 0; k < T_STEPS; ++k) {
    const float* vp = v2 + (size_t)n * ROWSTRIDE + k * C_DIM + h * HD;
    #pragma unroll
    for (int c = 0; c < HD; ++c) out[c] += sc[k] * vp[c];
  }
  float* op = o + (size_t)n * ROWSTRIDE + qi * C_DIM + h * HD;
  #pragma unroll
  for (int c = 0; c < HD; ++c) op[c] = out[c] * inv;
}

// ---------------- LayerNorm(X + R) * g + b, rows of 64 ----------------
__global__ void ln_res(const float* __restrict__ X, const float* __restrict__ R,
                       const float* __restrict__ g, const float* __restrict__ b,
                       float* __restrict__ O, int rows) {
  int row = blockIdx.x * (blockDim.x >> 5) + (threadIdx.x >> 5);
  if (row >= rows) return;
  int lane = threadIdx.x & 31;
  size_t base = (size_t)row * C_DIM;
  float x0 = X[base + lane]      + R[base + lane];
  float x1 = X[base + 32 + lane] + R[base + 32 + lane];
  float mean = red_sum32(x0 + x1) * (1.f / 64.f);
  float d0 = x0 - mean, d1 = x1 - mean;
  float var = red_sum32(d0 * d0 + d1 * d1) * (1.f / 64.f);
  float rstd = rsqrtf(var + 1e-5f);
  O[base + lane]      = d0 * rstd * g[lane]      + b[lane];
  O[base + 32 + lane] = d1 * rstd * g[32 + lane] + b[32 + lane];
}

// ---------------- elementwise ----------------
__global__ void add_node_embed(const float* __restrict__ X,
                               const float* __restrict__ Dn,
                               float* __restrict__ O) {   // O[n,t,c]=X+Dn[n,c]
  int i = blockIdx.x * blockDim.x + threadIdx.x;
  if (i >= NTC) return;
  int c = i % C_DIM;
  int n = i / ROWSTRIDE;
  O[i] = X[i] + Dn[n * C_DIM + c];
}

__global__ void add_time_embed(const float* __restrict__ X,
                               const float* __restrict__ temb,
                               const int* __restrict__ tptr,
                               float* __restrict__ O) {
  int i = blockIdx.x * blockDim.x + threadIdx.x;
  if (i >= NTC) return;
  int c  = i % C_DIM;
  int ti = (i / C_DIM) % T_STEPS;
  int tt = tptr[0];
  int idx = (TIME_NUM_ + tt + ti) % TIME_NUM_;
  O[i] = X[i] + temb[idx * C_DIM + c];
}

__global__ void gate_fuse(const float* __restrict__ gA, const float* __restrict__ gB,
                          const float* __restrict__ US, const float* __restrict__ XG,
                          float* __restrict__ O) {
  int i = blockIdx.x * blockDim.x + threadIdx.x;
  if (i >= NTC) return;
  float gv = 1.f / (1.f + __expf(-(gA[i] + gB[i])));
  O[i] = gv * US[i] + (1.f - gv) * XG[i];
}

// ---------------- host orchestration ----------------
extern "C" void kernel_launch(void* const* d_in, const int* in_sizes, int n_in,
                              void* d_out, int out_size, void* d_ws, size_t ws_size,
                              hipStream_t stream) {
  (void)in_sizes; (void)n_in; (void)out_size; (void)ws_size;
  const float* value   = (const float*)d_in[0];
  const float* key     = (const float*)d_in[1];
  const float* query   = (const float*)d_in[2];
  const float* adj     = (const float*)d_in[3];
  const float* s_embed_W = (const float*)d_in[4];
  const float* s_embed_b = (const float*)d_in[5];
  const float* sq_W = (const float*)d_in[6];
  const float* sk_W = (const float*)d_in[7];
  const float* sv_W = (const float*)d_in[8];
  const float* s_fc_W = (const float*)d_in[9];
  const float* s_fc_b = (const float*)d_in[10];
  const float* s_ln1_g = (const float*)d_in[11];
  const float* s_ln1_b = (const float*)d_in[12];
  const float* s_ln2_g = (const float*)d_in[13];
  const float* s_ln2_b = (const float*)d_in[14];
  const float* s_ff_W1 = (const float*)d_in[15];
  const float* s_ff_b1 = (const float*)d_in[16];
  const float* s_ff_W2 = (const float*)d_in[17];
  const float* s_ff_b2 = (const float*)d_in[18];
  const float* gcn_W1 = (const float*)d_in[19];
  const float* gcn_b1 = (const float*)d_in[20];
  const float* gcn_W2 = (const float*)d_in[21];
  const float* gcn_b2 = (const float*)d_in[22];
  const float* fs_W = (const float*)d_in[23];
  const float* fs_b = (const float*)d_in[24];
  const float* fg_W = (const float*)d_in[25];
  const float* fg_b = (const float*)d_in[26];
  const float* t_emb = (const float*)d_in[27];
  const float* tq_W = (const float*)d_in[28];
  const float* tk_W = (const float*)d_in[29];
  const float* tv_W = (const float*)d_in[30];
  const float* t_fc_W = (const float*)d_in[31];
  const float* t_fc_b = (const float*)d_in[32];
  const float* t_ln1_g = (const float*)d_in[33];
  const float* t_ln1_b = (const float*)d_in[34];
  const float* t_ln2_g = (const float*)d_in[35];
  const float* t_ln2_b = (const float*)d_in[36];
  const float* t_ff_W1 = (const float*)d_in[37];
  const float* t_ff_b1 = (const float*)d_in[38];
  const float* t_ff_W2 = (const float*)d_in[39];
  const float* t_ff_b2 = (const float*)d_in[40];
  const float* blk_ln1_g = (const float*)d_in[41];
  const float* blk_ln1_b = (const float*)d_in[42];
  const float* blk_ln2_g = (const float*)d_in[43];
  const float* blk_ln2_b = (const float*)d_in[44];
  const int*   tptr = (const int*)d_in[45];
  float* out = (float*)d_out;

  // workspace layout (floats), with lifetime-based reuse
  float* ws   = (float*)d_ws;
  float* D_S  = ws;                         //    76800
  float* h1   = ws + 76800;                 //  1843200   [N, T*128]
  float* H2   = ws + 1920000;               //  1843200
  float* XG1  = ws + 3763200;               //   921600
  float* X_G  = ws + 4684800;               //   921600
  float* qbuf = ws + 5606400;               //   921600
  float* q_   = ws + 6528000;               //   921600
  float* k_   = ws + 7449600;               //   921600
  float* v_   = ws + 8371200;               //   921600
  float* att  = ws + 9292800;               //   921600
  float* attf = ws + 10214400;              //   921600
  float* xbuf = ws + 11136000;              //   921600
  float* ffb  = ws + 12057600;              //  3686400   [NT, 256]
  float* ff   = ws + 15744000;              //   921600
  float* U_S  = ws + 16665600;              //   921600
  // reused regions (dead by the time they're reused):
  float* gA   = h1;                 float* gB   = h1 + NTC;
  float* s_out= H2;                 float* x1   = H2 + NTC;
  float* q2   = XG1;
  float* q2_  = q_;  float* k2_ = k_;  float* v2_ = v_;
  float* att2 = att; float* att2f = attf; float* x2 = xbuf;
  float* ff2  = ff;  float* t_out = U_S;

  // wide GEMM: 16x64 strip per wave (Nc % 64 == 0)
  auto gemm = [&](const float* A, const float* B, const float* bias, float* Dd,
                  int M, int K, int Nc, int bmod, int relu) {
    int tiles = (M / 16) * (Nc / 64);
    int blocks = (tiles + 7) / 8;   // 8 waves per 256-thread block
    gemm64_wmma<<<blocks, 256, 0, stream>>>(A, B, bias, Dd, M, K, Nc, bmod, relu);
  };
  // narrow GEMM for the Hd x Hd head projections (Nc == 16)
  auto gemm_hd = [&](const float* A, const float* B, float* Dd, int M) {
    int tiles = M / 16;
    int blocks = (tiles + 7) / 8;
    gemm16_wmma<<<blocks, 256, 0, stream>>>(A, B, nullptr, Dd, M, HD, HD, 1, 0);
  };
  const int EWB = (NTC + 255) / 256;     // 3600 blocks for elementwise
  const int LNB = (NT + 7) / 8;          // 1800 blocks for layernorm

  // ---- STransformer ----
  gemm(adj, s_embed_W, s_embed_b, D_S, N_NODES, N_NODES, C_DIM, C_DIM, 0);     // D_S
  add_node_embed<<<EWB, 256, 0, stream>>>(query, D_S, qbuf);                   // q = query + D_S
  gemm(query, gcn_W1, nullptr, h1, NT, C_DIM, C2_DIM, 1, 0);                   // h = query@W1
  gemm(adj, h1, gcn_b1, H2, N_NODES, N_NODES, T_STEPS * C2_DIM, C2_DIM, 1);    // relu(adj@h+b1)
  gemm(H2, gcn_W2, nullptr, XG1, NT, C2_DIM, C_DIM, 1, 0);                     // h@W2
  gemm(adj, XG1, gcn_b2, X_G, N_NODES, N_NODES, T_STEPS * C_DIM, C_DIM, 0);    // adj@..+b2
  gemm_hd(qbuf,  sq_W, q_, NT * H_HEADS);                                      // per-head projections
  gemm_hd(key,   sk_W, k_, NT * H_HEADS);
  gemm_hd(value, sv_W, v_, NT * H_HEADS);
  spatial_flash_attn<<<T_STEPS * H_HEADS * (N_NODES / 16), 32, 0, stream>>>(q_, k_, v_, att);
  gemm(att, s_fc_W, s_fc_b, attf, NT, C_DIM, C_DIM, C_DIM, 0);
  ln_res<<<LNB, 256, 0, stream>>>(attf, qbuf, s_ln1_g, s_ln1_b, xbuf, NT);     // x
  gemm(xbuf, s_ff_W1, s_ff_b1, ffb, NT, C_DIM, FC_DIM, FC_DIM, 1);
  gemm(ffb,  s_ff_W2, s_ff_b2, ff, NT, FC_DIM, C_DIM, C_DIM, 0);
  ln_res<<<LNB, 256, 0, stream>>>(ff, xbuf, s_ln2_g, s_ln2_b, U_S, NT);        // U_S
  gemm(U_S, fs_W, fs_b, gA, NT, C_DIM, C_DIM, C_DIM, 0);
  gemm(X_G, fg_W, fg_b, gB, NT, C_DIM, C_DIM, C_DIM, 0);
  gate_fuse<<<EWB, 256, 0, stream>>>(gA, gB, U_S, X_G, s_out);
  ln_res<<<LNB, 256, 0, stream>>>(s_out, query, blk_ln1_g, blk_ln1_b, x1, NT); // x1

  // ---- TTransformer ----
  add_time_embed<<<EWB, 256, 0, stream>>>(x1, t_emb, tptr, q2);                // q2 = x1 + D_T
  gemm_hd(q2, tq_W, q2_, NT * H_HEADS);
  gemm_hd(x1, tk_W, k2_, NT * H_HEADS);
  gemm_hd(x1, tv_W, v2_, NT * H_HEADS);
  temporal_attn<<<(N_NODES * H_HEADS * T_STEPS + 255) / 256, 256, 0, stream>>>(q2_, k2_, v2_, att2);
  gemm(att2, t_fc_W, t_fc_b, att2f, NT, C_DIM, C_DIM, C_DIM, 0);
  ln_res<<<LNB, 256, 0, stream>>>(att2f, q2, t_ln1_g, t_ln1_b, x2, NT);        // x2
  gemm(x2,  t_ff_W1, t_ff_b1, ffb, NT, C_DIM, FC_DIM, FC_DIM, 1);
  gemm(ffb, t_ff_W2, t_ff_b2, ff2, NT, FC_DIM, C_DIM, C_DIM, 0);
  ln_res<<<LNB, 256, 0, stream>>>(ff2, x2, t_ln2_g, t_ln2_b, t_out, NT);       // t_out
  ln_res<<<LNB, 256, 0, stream>>>(t_out, x1, blk_ln2_g, blk_ln2_b, out, NT);   // final
}